// GeneratorArchC_9131100471500
// MI455X (gfx1250) — compile-verified
//
#include <hip/hip_runtime.h>
#include <hip/hip_bf16.h>

typedef __attribute__((ext_vector_type(2))) float v2f;
typedef __attribute__((ext_vector_type(8))) float v8f;

#define NQ  6
#define DIM 64
#define NL  3

__device__ __forceinline__ int insert_zero(int v, int pos) {
  int m = (1 << pos) - 1;
  return ((v & ~m) << 1) | (v & m);
}

// ------------------------------------------------------------------
// Setup kernel (1 block x 1024 threads): build the folded-unitary
// matrices B_re/B_im (64x64) from `weights`.
//   B[j][k] = U[k][j] * (-i)^popcount(j)
// The full 64-column statevector bank lives in LDS:
//   ldsRe[amp*64 + col], col t = circuit applied to basis e_t.
// Each gate = disjoint pair updates, one (CNOT) or two (RY) per
// thread, __syncthreads() between gates.
// Wire w <-> bit position (5 - w) of the flat state index (C-order).
// Output packed as (j/2, k, j&1) so a WMMA lane fetches its
// B-fragment K-pair with one float2 load.
// ------------------------------------------------------------------
__global__ __launch_bounds__(1024) void qnn_build_B(
    const float* __restrict__ weights,
    float* __restrict__ Bre,
    float* __restrict__ Bim) {
  __shared__ float ldsRe[DIM * DIM];
  __shared__ float ldsIm[DIM * DIM];

  int tid = threadIdx.x;        // 0..1023
  int t   = tid & 63;           // column (basis state)
  int p   = tid >> 6;           // 0..15 work index

  // init: identity columns
  for (int i = tid; i < DIM * DIM; i += 1024) {
    int j = i >> 6, col = i & 63;
    ldsRe[i] = (j == col) ? 1.f : 0.f;
    ldsIm[i] = 0.f;
  }
  __syncthreads();

  for (int l = 0; l < NL; ++l) {
    // CNOT ring: control w, target (w+1)%6
    for (int w = 0; w < NQ; ++w) {
      int cpos = NQ - 1 - w;
      int tpos = NQ - 1 - ((w + 1) % NQ);
      int b1 = cpos < tpos ? cpos : tpos;
      int b2 = cpos < tpos ? tpos : cpos;
      // 16 pairs per column * 64 columns = 1024 -> one per thread
      int j  = insert_zero(insert_zero(p, b1), b2) | (1 << cpos);
      int j2 = j | (1 << tpos);
      int a0 = j * DIM + t, a1 = j2 * DIM + t;
      float r0 = ldsRe[a0], i0 = ldsIm[a0];
      float r1 = ldsRe[a1], i1 = ldsIm[a1];
      ldsRe[a0] = r1; ldsIm[a0] = i1;
      ldsRe[a1] = r0; ldsIm[a1] = i0;
      __syncthreads();
    }
    // RY(weights[l][w]) on each wire (real 2x2 rotation)
    for (int w = 0; w < NQ; ++w) {
      int pos = NQ - 1 - w;
      float th = 0.5f * weights[l * NQ + w];
      float cth = cosf(th), sth = sinf(th);
      // 32 pairs per column * 64 columns = 2048 -> two per thread
#pragma unroll
      for (int half = 0; half < 2; ++half) {
        int pp = p + 16 * half;                 // 0..31
        int j  = insert_zero(pp, pos);
        int j2 = j | (1 << pos);
        int a0 = j * DIM + t, a1 = j2 * DIM + t;
        float ar = ldsRe[a0], ai = ldsIm[a0];
        float br = ldsRe[a1], bi = ldsIm[a1];
        ldsRe[a0] = cth * ar - sth * br;  ldsIm[a0] = cth * ai - sth * bi;
        ldsRe[a1] = sth * ar + cth * br;  ldsIm[a1] = sth * ai + cth * bi;
      }
      __syncthreads();
    }
  }

  // apply (-i)^popcount(col) and store packed:
  //   B[t][k] = psi_t[k] * phase(t),  idx = ((t>>1)*64 + k)*2 + (t&1)
  for (int e = tid; e < DIM * DIM; e += 1024) {
    int col = e & 63, k = e >> 6;
    float vr = ldsRe[k * DIM + col];
    float vi = ldsIm[k * DIM + col];
    int pc = __popc(col) & 3;
    // pc==0: ( vr,  vi)   pc==1: ( vi, -vr)
    // pc==2: (-vr, -vi)   pc==3: (-vi,  vr)
    bool swp = pc & 1;
    float sR = (pc == 2 || pc == 3) ? -1.f : 1.f;
    float sI = (pc == 1 || pc == 2) ? -1.f : 1.f;
    float br = swp ? (sR * vi) : (sR * vr);
    float bi = swp ? (sI * vr) : (sI * vi);
    int idx = ((col >> 1) * DIM + k) * 2 + (col & 1);
    Bre[idx] = br;
    Bim[idx] = bi;
  }
}

// ------------------------------------------------------------------
// Main kernel: per wave, 16 samples.  Y = R @ B (re and im) via
// V_WMMA_F32_16X16X4_F32, then P = Yre^2 + Yim^2, then signed
// reductions give the six <Z_w> per sample.
// ------------------------------------------------------------------
__global__ __launch_bounds__(256) void qnn_main(
    const float* __restrict__ x,
    const float* __restrict__ gBre,
    const float* __restrict__ gBim,
    float* __restrict__ out,
    int nsamples) {
  __shared__ __align__(16) float ldsBre[DIM * DIM];
  __shared__ __align__(16) float ldsBim[DIM * DIM];

  int tid = threadIdx.x;
  // cooperative load of the two 16 KB B matrices into LDS
  for (int i = tid; i < DIM * DIM / 4; i += 256) {
    ((float4*)ldsBre)[i] = ((const float4*)gBre)[i];
    ((float4*)ldsBim)[i] = ((const float4*)gBim)[i];
  }
  __syncthreads();

  int wave = tid >> 5;
  int lane = tid & 31;
  int c = lane & 15;        // WMMA column / sample-row selector
  int h = lane >> 4;        // half-wave: selects K-pair of fragments

  long s0 = ((long)blockIdx.x * 8 + wave) * 16;   // first sample of this wave
  if (s0 >= nsamples) return;                      // wave-uniform (EXEC stays all-1)

  // ---- per-sample encoding factors: fc/fs indexed by BIT position p = 5-w
  long sl = s0 + c; if (sl > nsamples - 1) sl = nsamples - 1;
  float fc[NQ], fs[NQ];
#pragma unroll
  for (int w = 0; w < NQ; ++w) {
    float th = 0.5f * x[sl * NQ + w];
    float sth, cth;
    __sincosf(th, &sth, &cth);
    fc[NQ - 1 - w] = cth;
    fs[NQ - 1 - w] = sth;
  }

  // A-fragment element j = 4*q + 2*h + e  (bit0=e, bit1=h, bits2..5=q)
  float f1h = h ? fs[1] : fc[1];
  float pe0 = f1h * fc[0];
  float pe1 = f1h * fs[0];
  float hihi[4], lohi[4];
#pragma unroll
  for (int q2 = 0; q2 < 4; ++q2) {
    hihi[q2] = ((q2 & 2) ? fs[5] : fc[5]) * ((q2 & 1) ? fs[4] : fc[4]);
    lohi[q2] = ((q2 & 2) ? fs[3] : fc[3]) * ((q2 & 1) ? fs[2] : fc[2]);
  }

  v8f accRe[4] = {}, accIm[4] = {};
  const v2f* B2re = (const v2f*)ldsBre;
  const v2f* B2im = (const v2f*)ldsBim;

#pragma unroll
  for (int kk = 0; kk < DIM; kk += 4) {
    int q = kk >> 2;
    float hi = hihi[q >> 2] * lohi[q & 3];
    v2f a;
    a.x = hi * pe0;            // R[s, kk + 2h]
    a.y = hi * pe1;            // R[s, kk + 2h + 1]
    int jp = (kk >> 1) + h;    // packed j-pair row
#pragma unroll
    for (int n = 0; n < 4; ++n) {
      int k = n * 16 + c;
      v2f br = B2re[jp * DIM + k];
      v2f bi = B2im[jp * DIM + k];
      accRe[n] = __builtin_amdgcn_wmma_f32_16x16x4_f32(
          false, a, false, br, (short)0, accRe[n], false, false);
      accIm[n] = __builtin_amdgcn_wmma_f32_16x16x4_f32(
          false, a, false, bi, (short)0, accIm[n], false, false);
    }
  }

  // probabilities, tile n holds k = 16n + c for sample M = g + 8h
  v8f P[4];
#pragma unroll
  for (int n = 0; n < 4; ++n) P[n] = accRe[n] * accRe[n] + accIm[n] * accIm[n];

#pragma unroll
  for (int g = 0; g < 8; ++g) {
    float p0 = P[0][g], p1 = P[1][g], p2 = P[2][g], p3 = P[3][g];
    float sall = (p0 + p1) + (p2 + p3);
    float res[6];
    res[0] = (p0 + p1) - (p2 + p3);          // w=0: bit5(k) = bit1(n)
    res[1] = (p0 + p2) - (p1 + p3);          // w=1: bit4(k) = bit0(n)
    res[2] = (c & 8) ? -sall : sall;         // w=2: bit3(k) = bit3(c)
    res[3] = (c & 4) ? -sall : sall;
    res[4] = (c & 2) ? -sall : sall;
    res[5] = (c & 1) ? -sall : sall;
#pragma unroll
    for (int w = 0; w < NQ; ++w) {
      float v = res[w];
      v += __shfl_xor(v, 1, 32);
      v += __shfl_xor(v, 2, 32);
      v += __shfl_xor(v, 4, 32);
      v += __shfl_xor(v, 8, 32);
      res[w] = v;
    }
    if (c == 0) {
      long ss = s0 + g + 8 * h;
      if (ss < nsamples) {
#pragma unroll
        for (int w = 0; w < NQ; ++w) out[ss * NQ + w] = res[w];
      }
    }
  }
}

// ------------------------------------------------------------------
extern "C" void kernel_launch(void* const* d_in, const int* in_sizes, int n_in,
                              void* d_out, int out_size, void* d_ws, size_t ws_size,
                              hipStream_t stream) {
  const float* x       = (const float*)d_in[0];
  const float* weights = (const float*)d_in[1];
  float* out = (float*)d_out;

  float* Bre = (float*)d_ws;             // 64*64 floats
  float* Bim = Bre + DIM * DIM;          // 64*64 floats (32 KB total)

  qnn_build_B<<<dim3(1), dim3(1024), 0, stream>>>(weights, Bre, Bim);

  int nsamples = in_sizes[0] / NQ;
  int blocks = (nsamples + 127) / 128;   // 128 samples per block (8 waves x 16)
  qnn_main<<<dim3(blocks), dim3(256), 0, stream>>>(x, Bre, Bim, out, nsamples);
}